// Net_12403865551680
// MI455X (gfx1250) — compile-verified
//
#include <hip/hip_runtime.h>
#include <hip/hip_bf16.h>

typedef __attribute__((ext_vector_type(2))) float v2f;
typedef __attribute__((ext_vector_type(8))) float v8f;

#define NG 1024

// ---------------------------------------------------------------------------
// fp32 WMMA GEMM: C[M,N] = A[M,K] @ B[K,N]  (row-major with leading dims).
// One wave computes a 64x16 strip = 4 stacked 16x16 WMMA tiles, sharing the
// B fragment across the 4 V_WMMA_F32_16X16X4_F32 issues per k-step.
// Ragged M/N edges handled by clamping load ADDRESSES (loads stay
// unconditional -> no exec-mask branches in the hot loop); the resulting
// garbage rows/cols are simply not stored. Requires K % 4 == 0 (true for all
// call sites: 128, 100, 640, 128).
// A frag (16x4): lanes 0-15 hold M=0..15 {K=k0,k0+1}; lanes 16-31 {k0+2,k0+3}
// B frag (4x16): N=lane%16; low half K=k0,k0+1, high half K=k0+2,k0+3
// C frag (16x16): VGPR v: lanes 0-15 M=v, lanes 16-31 M=v+8, N=lane%16
// ---------------------------------------------------------------------------
__global__ void gemm_f32_wmma(const float* __restrict__ A,
                              const float* __restrict__ B,
                              float* __restrict__ C,
                              int M, int K, int N, int lda, int ldb, int ldc) {
  int wave = (blockIdx.x * blockDim.x + threadIdx.x) >> 5;
  int lane = threadIdx.x & 31;
  int tilesM = (M + 63) >> 6;                 // 64 rows per wave
  int tilesN = (N + 15) >> 4;
  if (wave >= tilesM * tilesN) return;        // wave-uniform: EXEC all-1s below
  int mt = wave / tilesN;
  int nt = wave % tilesN;
  int m0 = mt << 6, n0 = nt << 4;
  int lrow  = lane & 15;
  int khalf = (lane >> 4) << 1;               // 0 or 2

  int nl = n0 + lrow; if (nl > N - 1) nl = N - 1;   // clamped column address
  const float* brow = B + nl;
  int mA = m0 + lrow;
  const float* a0 = A + (size_t)(mA      < M ? mA      : M - 1) * lda;
  const float* a1 = A + (size_t)(mA + 16 < M ? mA + 16 : M - 1) * lda;
  const float* a2 = A + (size_t)(mA + 32 < M ? mA + 32 : M - 1) * lda;
  const float* a3 = A + (size_t)(mA + 48 < M ? mA + 48 : M - 1) * lda;

  v8f c0 = {}, c1 = {}, c2 = {}, c3 = {};
  for (int k0 = 0; k0 < K; k0 += 4) {
    int ka = k0 + khalf;                      // even -> v2f load is 8B aligned
    v2f b;
    b.x = brow[(size_t)ka * ldb];
    b.y = brow[(size_t)(ka + 1) * ldb];
    v2f x0 = *(const v2f*)(a0 + ka);
    v2f x1 = *(const v2f*)(a1 + ka);
    v2f x2 = *(const v2f*)(a2 + ka);
    v2f x3 = *(const v2f*)(a3 + ka);
    c0 = __builtin_amdgcn_wmma_f32_16x16x4_f32(false, x0, false, b, (short)0, c0, false, false);
    c1 = __builtin_amdgcn_wmma_f32_16x16x4_f32(false, x1, false, b, (short)0, c1, false, false);
    c2 = __builtin_amdgcn_wmma_f32_16x16x4_f32(false, x2, false, b, (short)0, c2, false, false);
    c3 = __builtin_amdgcn_wmma_f32_16x16x4_f32(false, x3, false, b, (short)0, c3, false, false);
  }

  if (n0 + lrow < N) {
    int mhi = (lane >> 4) ? 8 : 0;
    size_t col = (size_t)(n0 + lrow);
#pragma unroll
    for (int v = 0; v < 8; ++v) {
      int mm = m0 + v + mhi;
      if (mm      < M) C[(size_t)mm * ldc + col]        = c0[v];
      if (mm + 16 < M) C[(size_t)(mm + 16) * ldc + col] = c1[v];
      if (mm + 32 < M) C[(size_t)(mm + 32) * ldc + col] = c2[v];
      if (mm + 48 < M) C[(size_t)(mm + 48) * ldc + col] = c3[v];
    }
  }
}

// Zero a contiguous float region.
__global__ void zero_f32(float* __restrict__ p, size_t n) {
  size_t i = (size_t)blockIdx.x * blockDim.x + threadIdx.x;
  if (i < n) p[i] = 0.0f;
}

// One wave per edge: psum1[dst] += P1[src] (100 dims, row stride 112), deg[dst] += 1
__global__ void scatter1(const float* __restrict__ P1, const int* __restrict__ src,
                         const int* __restrict__ dst, float* __restrict__ psum,
                         float* __restrict__ deg, int nEdges) {
  int tid = blockIdx.x * blockDim.x + threadIdx.x;
  int e = tid >> 5, lane = tid & 31;
  if (e >= nEdges) return;
  int s = src[e], d = dst[e];
  const float* ps = P1 + (size_t)s * 112;
  float* pd = psum + (size_t)d * 112;
#pragma unroll
  for (int b = 0; b < 4; ++b) {
    int j = b * 32 + lane;
    if (j < 100) atomicAdd(&pd[j], ps[j]);
  }
  if (lane == 0) atomicAdd(&deg[d], 1.0f);
}

// One wave per edge: psum2[dst] += P2[src] (20 dims)
__global__ void scatter2(const float* __restrict__ P2, const int* __restrict__ src,
                         const int* __restrict__ dst, float* __restrict__ psum,
                         int nEdges) {
  int tid = blockIdx.x * blockDim.x + threadIdx.x;
  int e = tid >> 5, lane = tid & 31;
  if (e >= nEdges) return;
  int s = src[e], d = dst[e];
  if (lane < 20)
    atomicAdd(&psum[(size_t)d * 20 + lane], P2[(size_t)s * 20 + lane]);
}

// h1 = relu(where(deg>0, psum1/deg, P1) + b1); in-place over psum1; pad cols 100..111 = 0
__global__ void combine1(float* __restrict__ h, const float* __restrict__ P1,
                         const float* __restrict__ deg, const float* __restrict__ b1,
                         int nNodes) {
  int tid = blockIdx.x * blockDim.x + threadIdx.x;
  int i = tid / 112, j = tid % 112;
  if (i >= nNodes) return;
  size_t idx = (size_t)i * 112 + j;
  float out = 0.0f;
  if (j < 100) {
    float dg = deg[i];
    float v = (dg > 0.0f) ? (h[idx] / dg) : P1[idx];
    out = v + b1[j];
    out = out > 0.0f ? out : 0.0f;
  }
  h[idx] = out;
}

// h2 = relu(where(deg>0, psum2/deg, P2) + b2); in-place over psum2 (20 dims)
__global__ void combine2(float* __restrict__ h, const float* __restrict__ P2,
                         const float* __restrict__ deg, const float* __restrict__ b2,
                         int nNodes) {
  int tid = blockIdx.x * blockDim.x + threadIdx.x;
  if (tid >= nNodes * 20) return;
  int i = tid / 20, j = tid % 20;
  size_t idx = (size_t)i * 20 + j;
  float dg = deg[i];
  float v = (dg > 0.0f) ? (h[idx] / dg) : P2[idx];
  v += b2[j];
  h[idx] = v > 0.0f ? v : 0.0f;
}

// One wave per node: per-graph sums + counts
__global__ void pool(const float* __restrict__ h2, const int* __restrict__ n2g,
                     float* __restrict__ gsum, float* __restrict__ gcnt, int nNodes) {
  int tid = blockIdx.x * blockDim.x + threadIdx.x;
  int i = tid >> 5, lane = tid & 31;
  if (i >= nNodes) return;
  int g = n2g[i];
  if (lane < 20) atomicAdd(&gsum[(size_t)g * 20 + lane], h2[(size_t)i * 20 + lane]);
  if (lane == 0) atomicAdd(&gcnt[g], 1.0f);
}

// fused[g, j*32+s] = (gsum[g,j]/max(cnt,1)) * self_feat[g,s]
__global__ void fuse(const float* __restrict__ gsum, const float* __restrict__ gcnt,
                     const float* __restrict__ sf, float* __restrict__ fused) {
  int g = blockIdx.x;
  float cnt = gcnt[g];
  cnt = cnt > 1.0f ? cnt : 1.0f;
  for (int t = threadIdx.x; t < 640; t += blockDim.x) {
    int j = t >> 5, s = t & 31;
    fused[(size_t)g * 640 + t] = (gsum[(size_t)g * 20 + j] / cnt) * sf[(size_t)g * 32 + s];
  }
}

// Batch-norm stats: one wave per column, biased stats over `rows` rows.
__global__ void bnstats(const float* __restrict__ x, float* __restrict__ mu,
                        float* __restrict__ rinv, int rows, int cols) {
  int tid = blockIdx.x * blockDim.x + threadIdx.x;
  int c = tid >> 5, lane = tid & 31;
  if (c >= cols) return;
  float s = 0.0f, s2 = 0.0f;
  for (int r = lane; r < rows; r += 32) {
    float v = x[(size_t)r * cols + c];
    s += v; s2 += v * v;
  }
#pragma unroll
  for (int o = 16; o > 0; o >>= 1) {
    s  += __shfl_xor(s, o, 32);
    s2 += __shfl_xor(s2, o, 32);
  }
  if (lane == 0) {
    float m = s / (float)rows;
    float var = s2 / (float)rows - m * m;
    mu[c] = m;
    rinv[c] = rsqrtf(var + 1e-5f);
  }
}

// y = relu((x - mu) * rinv * gamma + beta)
__global__ void bnapply(const float* __restrict__ x, const float* __restrict__ mu,
                        const float* __restrict__ rinv, const float* __restrict__ gamma,
                        const float* __restrict__ beta, float* __restrict__ y,
                        int rows, int cols) {
  int tid = blockIdx.x * blockDim.x + threadIdx.x;
  if (tid >= rows * cols) return;
  int c = tid % cols;
  float v = (x[tid] - mu[c]) * rinv[c] * gamma[c] + beta[c];
  y[tid] = v > 0.0f ? v : 0.0f;
}

// out[g,o] = sum_k y2[g,k]*Wf3[k,o] + bf3[o]   (1024x32x8 -> trivially small)
__global__ void final_fc(const float* __restrict__ y2, const float* __restrict__ W,
                         const float* __restrict__ b, float* __restrict__ out) {
  int tid = blockIdx.x * blockDim.x + threadIdx.x;
  if (tid >= NG * 8) return;
  int g = tid >> 3, o = tid & 7;
  float acc = b[o];
#pragma unroll
  for (int k = 0; k < 32; ++k) acc += y2[(size_t)g * 32 + k] * W[k * 8 + o];
  out[tid] = acc;
}

static inline void launch_gemm(const float* A, const float* B, float* C,
                               int M, int K, int N, int lda, int ldb, int ldc,
                               hipStream_t s) {
  int waves = ((M + 63) / 64) * ((N + 15) / 16);
  int blocks = (waves + 3) / 4;              // 4 waves (128 threads) per block
  gemm_f32_wmma<<<blocks, 128, 0, s>>>(A, B, C, M, K, N, lda, ldb, ldc);
}

extern "C" void kernel_launch(void* const* d_in, const int* in_sizes, int n_in,
                              void* d_out, int out_size, void* d_ws, size_t ws_size,
                              hipStream_t stream) {
  const float* feat  = (const float*)d_in[0];   // [N,128]
  const float* sfeat = (const float*)d_in[1];   // [G,32]
  const int*   src   = (const int*)  d_in[2];   // [E]
  const int*   dst   = (const int*)  d_in[3];   // [E]
  const int*   n2g   = (const int*)  d_in[4];   // [N]
  const float* W1  = (const float*)d_in[5];     // [128,100]
  const float* b1  = (const float*)d_in[6];
  const float* W2  = (const float*)d_in[7];     // [100,20]
  const float* b2  = (const float*)d_in[8];
  const float* Wf1 = (const float*)d_in[9];     // [640,128]
  const float* g1  = (const float*)d_in[11];
  const float* be1 = (const float*)d_in[12];
  const float* Wf2 = (const float*)d_in[13];    // [128,32]
  const float* g2  = (const float*)d_in[15];
  const float* be2 = (const float*)d_in[16];
  const float* Wf3 = (const float*)d_in[17];    // [32,8]
  const float* bf3 = (const float*)d_in[18];
  float* out = (float*)d_out;                   // [1024,8]
  // bf1/bf2 intentionally unused: BN's mean subtraction cancels a constant
  // pre-BN bias exactly, so skipping them is mathematically identical.

  const int nNodes = in_sizes[0] / 128;         // 100000
  const int nEdges = in_sizes[2];               // 1600000
  const int nG     = in_sizes[1] / 32;          // 1024

  // Workspace layout (floats). Zero-needed accumulators are contiguous.
  float* ws = (float*)d_ws;
  size_t oP1  = 0;                              // P1  [N,112] (100 cols, padded)
  size_t oPS1 = oP1  + (size_t)nNodes * 112;    // psum1 -> h1 (in place) [N,112]
  size_t oDeg = oPS1 + (size_t)nNodes * 112;    // deg [N]
  size_t oPS2 = oDeg + (size_t)nNodes;          // psum2 -> h2 (in place) [N,20]
  size_t oGS  = oPS2 + (size_t)nNodes * 20;     // gsum [G,20]
  size_t oGC  = oGS  + (size_t)nG * 20;         // gcnt [G]
  size_t zN   = (oGC + nG) - oPS1;              // contiguous zero block
  size_t oP2  = oGC  + (size_t)nG;              // P2 [N,20]
  size_t oFus = oP2  + (size_t)nNodes * 20;     // fused [G,640]
  size_t oX1  = oFus + (size_t)nG * 640;        // x1 [G,128]
  size_t oMu1 = oX1  + (size_t)nG * 128;        // 128
  size_t oRi1 = oMu1 + 128;                     // 128
  size_t oY1  = oRi1 + 128;                     // y1 [G,128]
  size_t oX2  = oY1  + (size_t)nG * 128;        // x2 [G,32]
  size_t oMu2 = oX2  + (size_t)nG * 32;         // 32
  size_t oRi2 = oMu2 + 32;                      // 32
  size_t oY2  = oRi2 + 32;                      // y2 [G,32]

  // 1) zero accumulators
  zero_f32<<<(int)((zN + 255) / 256), 256, 0, stream>>>(ws + oPS1, zN);

  // 2) P1 = feat @ W1   [N,128]x[128,100] -> [N,100] (ldc=112)
  launch_gemm(feat, W1, ws + oP1, nNodes, 128, 100, 128, 100, 112, stream);

  // 3) edge scatter of projected features + degree
  scatter1<<<(nEdges * 32 + 255) / 256, 256, 0, stream>>>(
      ws + oP1, src, dst, ws + oPS1, ws + oDeg, nEdges);

  // 4) h1 = relu(where(deg>0, psum1/deg, P1) + b1)
  combine1<<<(nNodes * 112 + 255) / 256, 256, 0, stream>>>(
      ws + oPS1, ws + oP1, ws + oDeg, b1, nNodes);

  // 5) P2 = h1 @ W2   [N,100]x[100,20] -> [N,20]
  launch_gemm(ws + oPS1, W2, ws + oP2, nNodes, 100, 20, 112, 20, 20, stream);

  // 6) scatter layer-2 messages
  scatter2<<<(nEdges * 32 + 255) / 256, 256, 0, stream>>>(
      ws + oP2, src, dst, ws + oPS2, nEdges);

  // 7) h2 = relu(where(deg>0, psum2/deg, P2) + b2)
  combine2<<<(nNodes * 20 + 255) / 256, 256, 0, stream>>>(
      ws + oPS2, ws + oP2, ws + oDeg, b2, nNodes);

  // 8) per-graph mean pooling
  pool<<<(nNodes * 32 + 255) / 256, 256, 0, stream>>>(
      ws + oPS2, n2g, ws + oGS, ws + oGC, nNodes);

  // 9) Kronecker fusion -> fused [G,640]
  fuse<<<nG, 256, 0, stream>>>(ws + oGS, ws + oGC, sfeat, ws + oFus);

  // 10) x1 = fused @ Wf1 ; BN ; relu
  launch_gemm(ws + oFus, Wf1, ws + oX1, nG, 640, 128, 640, 128, 128, stream);
  bnstats<<<(128 * 32 + 255) / 256, 256, 0, stream>>>(ws + oX1, ws + oMu1, ws + oRi1, nG, 128);
  bnapply<<<(nG * 128 + 255) / 256, 256, 0, stream>>>(
      ws + oX1, ws + oMu1, ws + oRi1, g1, be1, ws + oY1, nG, 128);

  // 11) x2 = y1 @ Wf2 ; BN ; relu
  launch_gemm(ws + oY1, Wf2, ws + oX2, nG, 128, 32, 128, 32, 32, stream);
  bnstats<<<(32 * 32 + 255) / 256, 256, 0, stream>>>(ws + oX2, ws + oMu2, ws + oRi2, nG, 32);
  bnapply<<<(nG * 32 + 255) / 256, 256, 0, stream>>>(
      ws + oX2, ws + oMu2, ws + oRi2, g2, be2, ws + oY2, nG, 32);

  // 12) out = y2 @ Wf3 + bf3
  final_fc<<<(nG * 8 + 255) / 256, 256, 0, stream>>>(ws + oY2, Wf3, bf3, out);
}